// DecoderRNNWrapped_65644280152285
// MI455X (gfx1250) — compile-verified
//
#include <hip/hip_runtime.h>

// MI455X / gfx1250 decoder-LSTM. FP32 WMMA (V_WMMA_F32_16X16X4_F32) keeps the
// argmax feedback chain numerically faithful to the fp32 reference while still
// running on the CDNA5 matrix pipe. Whole weight set (~148MB) is L2-resident.

typedef __attribute__((ext_vector_type(2))) float v2f;
typedef __attribute__((ext_vector_type(8))) float v8f;

#define LAYERS 2
#define BATCH  32
#define HID    512
#define VOCAB  32000
#define TLEN   128
#define G4     (4 * HID)          // 2048 gate width

// workspace layout (float offsets)
#define WS_HSTATE 0                                   // [L][B][H]   = 32768
#define WS_CSTATE (WS_HSTATE + LAYERS * BATCH * HID)  // [L][B][H]   = 32768
#define WS_XBUF   (WS_CSTATE + LAYERS * BATCH * HID)  // [B][H]      = 16384
#define WS_GATES  (WS_XBUF + BATCH * HID)             // [B][4H]     = 65536
#define WS_LOGITS (WS_GATES + BATCH * G4)             // [B][V]      = 1024000
#define WS_TOK    (WS_LOGITS + BATCH * VOCAB)         // int[B]

__device__ __forceinline__ v8f wmma_f32(v2f a, v2f b, v8f c) {
  // 8-arg form: (neg_a, A, neg_b, B, c_mod, C, reuse_a, reuse_b)
  return __builtin_amdgcn_wmma_f32_16x16x4_f32(false, a, false, b, (short)0, c,
                                               false, false);
}

__device__ __forceinline__ float sigm(float x) { return 1.0f / (1.0f + expf(-x)); }

// -------------------- init: h0 = encoder_hidden, c0 = 0, tok = SOS --------
__global__ void init_kernel(const float* __restrict__ enc, float* __restrict__ ws) {
  int i = blockIdx.x * blockDim.x + threadIdx.x;
  if (i < LAYERS * BATCH * HID) {
    ws[WS_HSTATE + i] = enc[i];
    ws[WS_CSTATE + i] = 0.0f;
  }
  if (i < BATCH) ((int*)(ws + WS_TOK))[i] = 1;  // SOS_TOKEN
}

// -------------------- K1: embed+ReLU + both LSTM cells (single block) -----
__global__ void __launch_bounds__(1024)
lstm_step_kernel(const float* __restrict__ emb, const float* __restrict__ W_ih,
                 const float* __restrict__ W_hh, const float* __restrict__ b_ih,
                 const float* __restrict__ b_hh, float* __restrict__ ws) {
  float* hstate = ws + WS_HSTATE;
  float* cstate = ws + WS_CSTATE;
  float* xbuf   = ws + WS_XBUF;
  float* gates  = ws + WS_GATES;
  const int* tok = (const int*)(ws + WS_TOK);

  const int tid = threadIdx.x;

  // embed + relu into xbuf
  for (int i = tid; i < BATCH * HID; i += blockDim.x) {
    int b = i >> 9, hh = i & (HID - 1);
    float x = emb[(size_t)tok[b] * HID + hh];
    xbuf[i] = x > 0.0f ? x : 0.0f;
  }
  __syncthreads();

  const int wave = tid >> 5;       // 32 waves
  const int lane = tid & 31;
  const int half = lane >> 4;      // 0: K=0,1  1: K=2,3
  const int lm   = lane & 15;
  const int k0   = half * 2;

  for (int l = 0; l < LAYERS; ++l) {
    const float* xs = (l == 0) ? xbuf : hstate;         // layer1 input = new h0
    const float* hs = hstate + l * BATCH * HID;
    const float* wi = W_ih + (size_t)l * G4 * HID;
    const float* wh = W_hh + (size_t)l * G4 * HID;

    // gates[b][j] = sum_k xs[b][k]*wi[j][k] + sum_k hs[b][k]*wh[j][k]
    // tiles: 2 M-tiles (batch) x 128 N-tiles (gates), K split x-half / h-half
    for (int tile = wave; tile < 2 * (G4 / 16); tile += 32) {
      const int mt = tile & 1;
      const int nt = tile >> 1;
      const int arow = mt * 16 + lm;
      const int n = nt * 16 + lm;
      const float* xa  = xs + (size_t)arow * HID;
      const float* ha  = hs + (size_t)arow * HID;
      const float* wib = wi + (size_t)n * HID;
      const float* whb = wh + (size_t)n * HID;
      v8f acc = {};
      for (int kk = 0; kk < HID / 4; ++kk) {
        int kb = kk * 4 + k0;
        v2f a = {xa[kb], xa[kb + 1]};
        v2f b = {wib[kb], wib[kb + 1]};
        acc = wmma_f32(a, b, acc);
      }
      for (int kk = 0; kk < HID / 4; ++kk) {
        int kb = kk * 4 + k0;
        v2f a = {ha[kb], ha[kb + 1]};
        v2f b = {whb[kb], whb[kb + 1]};
        acc = wmma_f32(a, b, acc);
      }
      for (int v = 0; v < 8; ++v) {
        int row = mt * 16 + v + half * 8;   // C/D layout: vgpr v -> M=v / v+8
        gates[row * G4 + n] = acc[v];
      }
    }
    __syncthreads();

    // elementwise LSTM cell (PyTorch gate order i,f,g,o)
    const float* bi = b_ih + l * G4;
    const float* bh = b_hh + l * G4;
    float* hl = hstate + l * BATCH * HID;
    float* cl = cstate + l * BATCH * HID;
    for (int i = tid; i < BATCH * HID; i += blockDim.x) {
      int b = i >> 9, hh = i & (HID - 1);
      const float* gr = gates + b * G4;
      float gi = gr[hh]           + bi[hh]           + bh[hh];
      float gf = gr[HID + hh]     + bi[HID + hh]     + bh[HID + hh];
      float gg = gr[2 * HID + hh] + bi[2 * HID + hh] + bh[2 * HID + hh];
      float go = gr[3 * HID + hh] + bi[3 * HID + hh] + bh[3 * HID + hh];
      float cn = sigm(gf) * cl[i] + sigm(gi) * tanhf(gg);
      cl[i] = cn;
      hl[i] = sigm(go) * tanhf(cn);
    }
    __syncthreads();
  }
}

// -------------------- K2: logits = h1 @ W_out^T + b_out (WMMA) ------------
__global__ void __launch_bounds__(256)
logits_kernel(const float* __restrict__ W_out, const float* __restrict__ b_out,
              float* __restrict__ ws) {
  const float* h1 = ws + WS_HSTATE + BATCH * HID;  // hstate[1]
  float* logits = ws + WS_LOGITS;

  const int wave = threadIdx.x >> 5;
  const int lane = threadIdx.x & 31;
  const int half = lane >> 4;
  const int lm   = lane & 15;
  const int k0   = half * 2;

  const int ntile = blockIdx.x * 8 + wave;  // [0, 2000) exactly
  const int n = ntile * 16 + lm;
  const float* wb  = W_out + (size_t)n * HID;
  const float* a0p = h1 + (size_t)lm * HID;
  const float* a1p = h1 + (size_t)(16 + lm) * HID;

  v8f c0 = {}, c1 = {};
  for (int kk = 0; kk < HID / 4; ++kk) {
    int kb = kk * 4 + k0;
    v2f bf = {wb[kb], wb[kb + 1]};     // shared B fragment for both M-tiles
    v2f a0 = {a0p[kb], a0p[kb + 1]};
    v2f a1 = {a1p[kb], a1p[kb + 1]};
    c0 = wmma_f32(a0, bf, c0);
    c1 = wmma_f32(a1, bf, c1);
  }
  float bias = b_out[n];
  for (int v = 0; v < 8; ++v) {
    int r = v + half * 8;
    logits[(size_t)r * VOCAB + n]        = c0[v] + bias;
    logits[(size_t)(16 + r) * VOCAB + n] = c1[v] + bias;
  }
}

// -------------------- K3: log_softmax + greedy argmax (1 block / row) -----
__global__ void __launch_bounds__(256)
softmax_argmax_kernel(float* __restrict__ ws, float* __restrict__ out, int t) {
  __shared__ float red[256];
  __shared__ int redi[256];
  const float* row = ws + WS_LOGITS + (size_t)blockIdx.x * VOCAB;
  int* tok = (int*)(ws + WS_TOK);
  const int tid = threadIdx.x;

  // pass 1: max + first-index argmax
  float m = -INFINITY; int am = 0;
  for (int v = tid; v < VOCAB; v += 256) {
    float x = row[v];
    if (x > m) { m = x; am = v; }
  }
  red[tid] = m; redi[tid] = am;
  __syncthreads();
  for (int s = 128; s > 0; s >>= 1) {
    if (tid < s) {
      float xo = red[tid + s]; int io = redi[tid + s];
      if (xo > red[tid] || (xo == red[tid] && io < redi[tid])) {
        red[tid] = xo; redi[tid] = io;
      }
    }
    __syncthreads();
  }
  const float rmax = red[0];
  const int rarg = redi[0];
  __syncthreads();

  // pass 2: sum exp
  float s = 0.0f;
  for (int v = tid; v < VOCAB; v += 256) s += expf(row[v] - rmax);
  red[tid] = s;
  __syncthreads();
  for (int st = 128; st > 0; st >>= 1) {
    if (tid < st) red[tid] += red[tid + st];
    __syncthreads();
  }
  const float lse = logf(red[0]) + rmax;

  // pass 3: write log-probs for this (b, t)
  float* o = out + (size_t)blockIdx.x * TLEN * VOCAB + (size_t)t * VOCAB;
  for (int v = tid; v < VOCAB; v += 256) o[v] = row[v] - lse;
  if (tid == 0) tok[blockIdx.x] = rarg;
}

// -------------------- finalize: hT, cT into d_out tail --------------------
__global__ void finalize_kernel(const float* __restrict__ ws, float* __restrict__ out) {
  int i = blockIdx.x * blockDim.x + threadIdx.x;
  const size_t base = (size_t)BATCH * TLEN * VOCAB;
  if (i < LAYERS * BATCH * HID) {
    out[base + i] = ws[WS_HSTATE + i];
    out[base + LAYERS * BATCH * HID + i] = ws[WS_CSTATE + i];
  }
}

extern "C" void kernel_launch(void* const* d_in, const int* in_sizes, int n_in,
                              void* d_out, int out_size, void* d_ws, size_t ws_size,
                              hipStream_t stream) {
  (void)in_sizes; (void)n_in; (void)out_size; (void)ws_size;
  const float* enc   = (const float*)d_in[0];  // [L,B,H]
  const float* emb   = (const float*)d_in[1];  // [V,H]
  const float* W_ih  = (const float*)d_in[2];  // [L,4H,H]
  const float* W_hh  = (const float*)d_in[3];  // [L,4H,H]
  const float* b_ih  = (const float*)d_in[4];  // [L,4H]
  const float* b_hh  = (const float*)d_in[5];  // [L,4H]
  const float* W_out = (const float*)d_in[6];  // [V,H]
  const float* b_out = (const float*)d_in[7];  // [V]
  float* out = (float*)d_out;
  float* ws  = (float*)d_ws;

  init_kernel<<<(LAYERS * BATCH * HID + 255) / 256, 256, 0, stream>>>(enc, ws);
  for (int t = 0; t < TLEN; ++t) {
    lstm_step_kernel<<<1, 1024, 0, stream>>>(emb, W_ih, W_hh, b_ih, b_hh, ws);
    logits_kernel<<<VOCAB / 16 / 8, 256, 0, stream>>>(W_out, b_out, ws);
    softmax_argmax_kernel<<<BATCH, 256, 0, stream>>>(ws, out, t);
  }
  finalize_kernel<<<(LAYERS * BATCH * HID + 255) / 256, 256, 0, stream>>>(ws, out);
}